// SeparableGateS2Activation_SwiGLU_Merge_42305427866195
// MI455X (gfx1250) — compile-verified
//
#include <hip/hip_runtime.h>

// ---------------------------------------------------------------------------
// SeparableGateS2Activation (SwiGLU merge) for MI455X / gfx1250.
//   inputs:        (N=4096, 49, 256) f32
//   scalars:       (N, 384)          f32
//   to_grid_mat:   (324, 49)         f32
//   from_grid_mat: (324, 49)         f32
//   out:           (N, 49, 128)      f32
//
// One block = one n.  8 waves, each wave owns a 16-channel slab end-to-end
// (channel dim is independent through both contractions) => no cross-wave
// exchange.  All matmuls are V_WMMA_F32_16X16X4_F32.
//
// GEMM1 fuses 3 row-tiles per K-loop (6 independent WMMA chains, B-fragment
// ds-loads amortized 3x).  GEMM2 fuses all 4 i-tiles (4 chains, Yt ds-loads
// amortized 4x).  All A-fragment guards replaced by address clamps: the B
// operand is zero-padded in K and out-of-range accumulator rows are masked
// only at the final store, so clamped (duplicated) A values contribute 0.
// ---------------------------------------------------------------------------

typedef __attribute__((ext_vector_type(2))) float v2f;
typedef __attribute__((ext_vector_type(8))) float v8f;

#define NCOEF 49          // SH coefficients (K of GEMM1, M of GEMM2)
#define KPAD 52           // 49 padded to multiple of 4
#define RES2 324          // 18*18 grid points
#define CTOT 128          // output channels
#define ITILE_STRIDE 32   // staged-input row stride in floats (16 ch + 16 ch)
#define ITILE_FLOATS (KPAD * ITILE_STRIDE)   // 1664 floats / wave
#define YTILE_FLOATS (RES2 * 16)             // 5184 floats / wave
#define SMEM_FLOATS (8 * (ITILE_FLOATS + YTILE_FLOATS))   // 219,136 bytes

__device__ __forceinline__ float sigmoidf_(float x) {
  return 1.0f / (1.0f + __expf(-x));
}

#define WMMA_F32(A, B, Cc) \
  __builtin_amdgcn_wmma_f32_16x16x4_f32(false, (A), false, (B), (short)0, (Cc), false, false)

__global__ __launch_bounds__(256, 1)
void s2act_swiglu_kernel(const float* __restrict__ X,   // (N,49,256)
                         const float* __restrict__ S,   // (N,384)
                         const float* __restrict__ G,   // (324,49) to_grid
                         const float* __restrict__ F,   // (324,49) from_grid
                         float* __restrict__ O)         // (N,49,128)
{
  extern __shared__ float smem[];
  const int n      = blockIdx.x;
  const int tid    = threadIdx.x;
  const int wave   = tid >> 5;        // wave32
  const int lane   = tid & 31;
  const int lane16 = lane & 15;
  const int lanehi = lane >> 4;       // 0 or 1
  const int cbase  = wave * 16;       // this wave's channel slab
  const int c      = cbase + lane16;  // this lane's channel (N index of WMMA)

  float* It = smem + wave * ITILE_FLOATS;                     // staged inputs 52x32
  float* Yt = smem + 8 * ITILE_FLOATS + wave * YTILE_FLOATS;  // Y column 324x16

  // ---- stage this wave's input columns (c, c+128) into LDS; zero-pad K->52 -
  const float* Xn = X + (size_t)n * (NCOEF * 2 * CTOT);
  const int ch     = c + CTOT * lanehi;        // lanes 0-15: c, lanes 16-31: c+128
  const int clocal = lane16 + 16 * lanehi;
  for (int i = 0; i < NCOEF; ++i)
    It[i * ITILE_STRIDE + clocal] = Xn[i * (2 * CTOT) + ch];
  #pragma unroll
  for (int i = NCOEF; i < KPAD; ++i)
    It[i * ITILE_STRIDE + clocal] = 0.0f;
  __syncthreads();

  // ---- GEMM1 + gate:  Y[pq,c] = (G@Xn)[pq,c] * (G@Xn)[pq,c+128] ------------
  // 21 row tiles processed as 7 groups of 3 (rows padded 324->336, clamped).
  for (int rtg = 0; rtg < 7; ++rtg) {
    const int rowbase = rtg * 48;
    const int r0 = rowbase      + lane16;
    const int r1 = rowbase + 16 + lane16;
    const int r2 = rowbase + 32 + lane16;
    const float* Ga = G + (size_t)(r0 < RES2 ? r0 : RES2 - 1) * NCOEF;
    const float* Gb = G + (size_t)(r1 < RES2 ? r1 : RES2 - 1) * NCOEF;
    const float* Gc = G + (size_t)(r2 < RES2 ? r2 : RES2 - 1) * NCOEF;
    v8f acc0 = {}, acc1 = {}, acc2 = {}, acc3 = {}, acc4 = {}, acc5 = {};
    for (int ks = 0; ks < 13; ++ks) {
      const int kb  = ks * 4 + 2 * lanehi;               // lane's K base
      const int kc0 = (kb     < NCOEF) ? kb     : NCOEF - 1;   // clamp (B is 0 there)
      const int kc1 = (kb + 1 < NCOEF) ? kb + 1 : NCOEF - 1;
      v2f b0, b1;                                        // B[K, N=lane16] from LDS
      b0.x = It[ kb      * ITILE_STRIDE + lane16];
      b0.y = It[(kb + 1) * ITILE_STRIDE + lane16];
      b1.x = It[ kb      * ITILE_STRIDE + 16 + lane16];
      b1.y = It[(kb + 1) * ITILE_STRIDE + 16 + lane16];
      v2f a0, a1, a2;                                    // A[M=row, K=kb..kb+1]
      a0.x = Ga[kc0]; a0.y = Ga[kc1];
      a1.x = Gb[kc0]; a1.y = Gb[kc1];
      a2.x = Gc[kc0]; a2.y = Gc[kc1];
      acc0 = WMMA_F32(a0, b0, acc0);
      acc1 = WMMA_F32(a0, b1, acc1);
      acc2 = WMMA_F32(a1, b0, acc2);
      acc3 = WMMA_F32(a1, b1, acc3);
      acc4 = WMMA_F32(a2, b0, acc4);
      acc5 = WMMA_F32(a2, b1, acc5);
    }
    // D layout: reg r -> (M = tile_base + r + 8*lanehi, N = lane16)
    #pragma unroll
    for (int r = 0; r < 8; ++r) {
      const int m0 = rowbase      + r + 8 * lanehi;
      const int m1 = rowbase + 16 + r + 8 * lanehi;
      const int m2 = rowbase + 32 + r + 8 * lanehi;
      if (m0 < RES2) Yt[m0 * 16 + lane16] = acc0[r] * acc1[r];
      if (m1 < RES2) Yt[m1 * 16 + lane16] = acc2[r] * acc3[r];
      if (m2 < RES2) Yt[m2 * 16 + lane16] = acc4[r] * acc5[r];
    }
  }
  __syncthreads();

  // ---- per-channel scalars: SwiGLU merge value + sigmoid gate --------------
  const float* Sn = S + (size_t)n * (3 * CTOT);
  const float sa    = Sn[c];
  const float sb    = Sn[CTOT + c];
  const float gate  = sigmoidf_(Sn[2 * CTOT + c]);
  const float oscal = sa * sigmoidf_(sa) * sb;   // silu(a)*b

  // ---- GEMM2:  O[i,c] = gate * sum_pq F[pq,i] * Y[pq,c]  (+ oscal at i==0) -
  // All 4 i-tiles fused in one K loop: shared Yt B-fragment, 4 WMMA chains.
  const float* F0 = F + ((     lane16) < NCOEF ? (     lane16) : NCOEF - 1);
  const float* F1 = F + ((16 + lane16) < NCOEF ? (16 + lane16) : NCOEF - 1);
  const float* F2 = F + ((32 + lane16) < NCOEF ? (32 + lane16) : NCOEF - 1);
  const float* F3 = F + ((48 + lane16) < NCOEF ? (48 + lane16) : NCOEF - 1);
  v8f oa0 = {}, oa1 = {}, oa2 = {}, oa3 = {};
  for (int ks = 0; ks < 81; ++ks) {            // K = 324 = 81*4 exactly
    const int p0 = ks * 4 + 2 * lanehi;        // pq base for this lane
    v2f b;                                     // B[K=pq, N=lane16] from LDS
    b.x = Yt[ p0      * 16 + lane16];
    b.y = Yt[(p0 + 1) * 16 + lane16];
    v2f a0, a1, a2, a3;                        // A[M=i, K=pq..pq+1] = F[pq, i]
    a0.x = F0[ p0      * NCOEF]; a0.y = F0[(p0 + 1) * NCOEF];
    a1.x = F1[ p0      * NCOEF]; a1.y = F1[(p0 + 1) * NCOEF];
    a2.x = F2[ p0      * NCOEF]; a2.y = F2[(p0 + 1) * NCOEF];
    a3.x = F3[ p0      * NCOEF]; a3.y = F3[(p0 + 1) * NCOEF];
    oa0 = WMMA_F32(a0, b, oa0);
    oa1 = WMMA_F32(a1, b, oa1);
    oa2 = WMMA_F32(a2, b, oa2);
    oa3 = WMMA_F32(a3, b, oa3);
  }

  float* On = O + (size_t)n * (NCOEF * CTOT) + c;
  #pragma unroll
  for (int r = 0; r < 8; ++r) {
    const int i0 = r + 8 * lanehi;             // i-tile 0
    float v0 = oa0[r] * gate;
    if (i0 == 0) v0 += oscal;
    On[i0 * CTOT] = v0;                        // i0 <= 15 < 49, always valid
    On[(16 + i0) * CTOT] = oa1[r] * gate;      // 16..31 valid
    On[(32 + i0) * CTOT] = oa2[r] * gate;      // 32..47 valid
    const int i3 = 48 + i0;
    if (i3 < NCOEF) On[i3 * CTOT] = oa3[r] * gate;   // only i=48
  }
}

extern "C" void kernel_launch(void* const* d_in, const int* in_sizes, int n_in,
                              void* d_out, int out_size, void* d_ws, size_t ws_size,
                              hipStream_t stream) {
  const float* X = (const float*)d_in[0];   // inputs        (N,49,256)
  const float* S = (const float*)d_in[1];   // scalars       (N,384)
  const float* G = (const float*)d_in[2];   // to_grid_mat   (324,49)
  const float* F = (const float*)d_in[3];   // from_grid_mat (324,49)
  float* O = (float*)d_out;                 // (N,49,128)

  const int N = in_sizes[0] / (NCOEF * 2 * CTOT);   // 4096
  const size_t shmem = (size_t)SMEM_FLOATS * sizeof(float);   // 219,136 B

  // gfx1250 WGP has 320 KB LDS; raise the dynamic-LDS cap for this kernel.
  (void)hipFuncSetAttribute((const void*)s2act_swiglu_kernel,
                            hipFuncAttributeMaxDynamicSharedMemorySize,
                            (int)shmem);

  hipLaunchKernelGGL(s2act_swiglu_kernel, dim3(N), dim3(256), shmem, stream,
                     X, S, G, F, O);
}